// DiscoveryNet_247
// MI455X (gfx1250) — compile-verified
//
#include <hip/hip_runtime.h>
#include <math.h>

typedef float v2f __attribute__((ext_vector_type(2)));
typedef float v8f __attribute__((ext_vector_type(8)));

#if __has_builtin(__builtin_amdgcn_tanhf)
#define TANHF(x) __builtin_amdgcn_tanhf(x)
#else
#define TANHF(x) tanhf(x)
#endif

// D = A(16x4 f32) * B(4x16 f32) + C(16x16 f32), wave32 WMMA
#define WMMA_F32(a, b, c) \
  __builtin_amdgcn_wmma_f32_16x16x4_f32(false, (a), false, (b), (short)0, (c), false, false)

static constexpr int NPART  = 1024;
static constexpr int JSPLIT = 8;            // blockIdx.y
static constexpr int WAVES  = 8;            // 256 threads
static constexpr int JPW    = (NPART / JSPLIT) / WAVES;   // 16 j's per wave

// Hidden-unit permutation: the value supplied to a WMMA B-operand at
// K-position kk = 4*kc + 2*H + v is the C-tile resident value
// X[kc>>2][2*(kc&3)+v] in lane-half H, i.e. hidden unit
//     R(kk) = 16*(kc>>2) + 8*H + 2*(kc&3) + v.
// Consuming A-fragments are loaded with rows permuted by R(kk) so the
// C->B re-layout between layers is a pure register pick (no lane movement).
__device__ __forceinline__ int permR(int kc, int H, int v) {
  return 16 * (kc >> 2) + 8 * H + 2 * (kc & 3) + v;
}

// Main kernel: block (ib, js) computes partial forces for particles
// [16*ib, 16*ib+16) over j in [js*128, js*128+128), writes to ws[js].
__global__ __launch_bounds__(256, 1) void forces_kernel(
    const float* __restrict__ pos, const float* __restrict__ W1,
    const float* __restrict__ b1,  const float* __restrict__ W2,
    const float* __restrict__ b2,  const float* __restrict__ W3,
    float* __restrict__ ws)
{
  __shared__ float sPos[NPART * 3];
  __shared__ float sRed[WAVES][16][3];

  for (int t = threadIdx.x; t < NPART * 3; t += blockDim.x) sPos[t] = pos[t];
  __syncthreads();

  const int lane = threadIdx.x & 31;
  const int wave = threadIdx.x >> 5;
  const int p    = lane & 15;        // pair column (N index of WMMA)
  const int H    = lane >> 4;        // lane half: 0 = lanes 0-15, 1 = lanes 16-31
  const bool lo  = (H == 0);

  const int ib = blockIdx.x;         // 0..63
  const int js = blockIdx.y;         // 0..7
  const int i0 = ib * 16;
  const int i  = i0 + p;

  // ---------------- loop-invariant weight fragments (registers) ----------------
  // A-frag layout (16x4 f32): lane L holds M=L%16, K = v + 2*(L/16) for vgpr v.
  // Layer1 fwd:  A = W1^T rows o=16mt+p, K = feat index (K=3 -> bias slot b1).
  v2f w1a[4];
  // Layer2 fwd:  A[mt][kc]: W2[permR][16mt+p]   (D[o,pair] = sum_k W2[k,o] h1[k,pair])
  // Layer2 bwd:  A[mt][kc]: W2[16mt+p][permR]   (D[k,pair] = sum_o W2[k,o] g2[o,pair])
  v2f w2f[4][16], w2b[4][16];
  // gfeat GEMM:  A[kc]: rows c=p (<3), K permuted: W1[p][permR]
  v2f w1bf[16];
  // Per-C-row constants (C row R = 16mt + 8H + r)
  float w3v[4][8];
  // b2 held as ready-to-use WMMA C tiles (loop-invariant v8f registers)
  v8f b2c[4];

#pragma unroll
  for (int mt = 0; mt < 4; ++mt) {
    const int o = 16 * mt + p;
    w1a[mt].x = lo ? W1[0 * 64 + o] : W1[2 * 64 + o];   // K=0 / K=2
    w1a[mt].y = lo ? W1[1 * 64 + o] : b1[o];            // K=1 / K=3 (bias)
#pragma unroll
    for (int kc = 0; kc < 16; ++kc) {
      const int r0 = permR(kc, H, 0);
      const int r1 = permR(kc, H, 1);
      w2f[mt][kc].x = W2[r0 * 64 + o];
      w2f[mt][kc].y = W2[r1 * 64 + o];
      w2b[mt][kc].x = W2[o * 64 + r0];
      w2b[mt][kc].y = W2[o * 64 + r1];
    }
#pragma unroll
    for (int r = 0; r < 8; ++r) {
      const int R = 16 * mt + 8 * H + r;
      w3v[mt][r] = W3[R];
      b2c[mt][r] = b2[R];
    }
  }
#pragma unroll
  for (int kc = 0; kc < 16; ++kc) {
    const int r0 = permR(kc, H, 0);
    const int r1 = permR(kc, H, 1);
    w1bf[kc].x = (p < 3) ? W1[p * 64 + r0] : 0.0f;
    w1bf[kc].y = (p < 3) ? W1[p * 64 + r1] : 0.0f;
  }

  // C-layout tile set -> B-frag for K-chunk kc: pure register pick.
  auto buildB = [&](const float (&X)[4][8], int kc) -> v2f {
    v2f b;
    b.x = X[kc >> 2][2 * (kc & 3) + 0];
    b.y = X[kc >> 2][2 * (kc & 3) + 1];
    return b;
  };

  const float pix = sPos[i * 3 + 0];
  const float piy = sPos[i * 3 + 1];
  const float piz = sPos[i * 3 + 2];
  float fx = 0.0f, fy = 0.0f, fz = 0.0f;

  const int jbeg = js * (NPART / JSPLIT) + wave * JPW;

#pragma unroll 1
  for (int j = jbeg; j < jbeg + JPW; ++j) {
    const float dx = pix - sPos[j * 3 + 0];
    const float dy = piy - sPos[j * 3 + 1];
    const float dz = piz - sPos[j * 3 + 2];
    const float d2 = dx * dx + dy * dy + dz * dz;
    const float s  = sqrtf(d2);
    const float d  = fmaxf(s, 0.05f);
    const float inv  = 1.0f / d;
    const float inv2 = inv * inv;

    // feat B-frag: rows [d, 1/d, 1/d^2, 1(bias)] (natural order; layer1 K=4)
    v2f bf;
    bf.x = lo ? d   : inv2;   // K=0 / K=2
    bf.y = lo ? inv : 1.0f;   // K=1 / K=3

    // ---- layer 1: h1 = tanh(W1^T feat + b1), stored at C row = unit ----
    float h1[4][8];
#pragma unroll
    for (int mt = 0; mt < 4; ++mt) {
      v8f z = {};
      v8f c = WMMA_F32(w1a[mt], bf, z);
#pragma unroll
      for (int r = 0; r < 8; ++r) h1[mt][r] = TANHF(c[r]);
    }

    // ---- layer 2 fwd: h2pre = W2^T h1 + b2 (bias via peeled C operand) ----
    v8f h2c[4];
    {
      const v2f bb0 = buildB(h1, 0);
#pragma unroll
      for (int mt = 0; mt < 4; ++mt) h2c[mt] = WMMA_F32(w2f[mt][0], bb0, b2c[mt]);
    }
#pragma unroll
    for (int kc = 1; kc < 16; ++kc) {
      const v2f bb = buildB(h1, kc);
#pragma unroll
      for (int mt = 0; mt < 4; ++mt) h2c[mt] = WMMA_F32(w2f[mt][kc], bb, h2c[mt]);
    }

    // ---- g2 = W3 * (1 - tanh(h2pre)^2) ----
    float g2[4][8];
#pragma unroll
    for (int mt = 0; mt < 4; ++mt)
#pragma unroll
      for (int r = 0; r < 8; ++r) {
        const float t = TANHF(h2c[mt][r]);
        g2[mt][r] = w3v[mt][r] * (1.0f - t * t);
      }

    // ---- layer 2 bwd: g1 = (W2 g2) * (1 - h1^2) (zero C peeled) ----
    v8f g1c[4];
    {
      const v2f bb0 = buildB(g2, 0);
#pragma unroll
      for (int mt = 0; mt < 4; ++mt) {
        v8f z = {};
        g1c[mt] = WMMA_F32(w2b[mt][0], bb0, z);
      }
    }
#pragma unroll
    for (int kc = 1; kc < 16; ++kc) {
      const v2f bb = buildB(g2, kc);
#pragma unroll
      for (int mt = 0; mt < 4; ++mt) g1c[mt] = WMMA_F32(w2b[mt][kc], bb, g1c[mt]);
    }
    float g1[4][8];
#pragma unroll
    for (int mt = 0; mt < 4; ++mt)
#pragma unroll
      for (int r = 0; r < 8; ++r)
        g1[mt][r] = g1c[mt][r] * (1.0f - h1[mt][r] * h1[mt][r]);

    // ---- gfeat = W1 g1 (rows 0..2 of a padded 16-row tile, zero C peeled) ----
    v8f gfc;
    {
      v8f z = {};
      gfc = WMMA_F32(w1bf[0], buildB(g1, 0), z);
    }
#pragma unroll
    for (int kc = 1; kc < 16; ++kc) {
      gfc = WMMA_F32(w1bf[kc], buildB(g1, kc), gfc);
    }
    // rows 0..2 live in lower half lanes (H=0, r=c); broadcast to pair's lanes
    const float gf0 = __shfl(gfc[0], p);
    const float gf1 = __shfl(gfc[1], p);
    const float gf2 = __shfl(gfc[2], p);

    // dv/ddist = gf.[1, -1/d^2, -2/d^3]; chain through clamp (zero if s < 0.05,
    // which also zeroes the j==i diagonal since s=0 there).
    const float vp    = gf0 - gf1 * inv2 - 2.0f * gf2 * inv2 * inv;
    const float scale = (s >= 0.05f) ? (vp / s) : 0.0f;
    fx -= scale * dx;
    fy -= scale * dy;
    fz -= scale * dz;
  }

  // both halves computed identical sums; keep lower half
  if (lo) {
    sRed[wave][p][0] = fx;
    sRed[wave][p][1] = fy;
    sRed[wave][p][2] = fz;
  }
  __syncthreads();

  if (threadIdx.x < 48) {
    const int pp = threadIdx.x / 3;
    const int c  = threadIdx.x % 3;
    float acc = 0.0f;
#pragma unroll
    for (int w = 0; w < WAVES; ++w) acc += sRed[w][pp][c];
    ws[js * (NPART * 3) + (i0 + pp) * 3 + c] = acc;   // deterministic slot
  }
}

// Deterministic final reduction over the JSPLIT partials.
__global__ void reduce_kernel(const float* __restrict__ ws, float* __restrict__ out) {
  const int t = blockIdx.x * blockDim.x + threadIdx.x;
  if (t < NPART * 3) {
    float a = 0.0f;
#pragma unroll
    for (int js = 0; js < JSPLIT; ++js) a += ws[js * (NPART * 3) + t];
    out[t] = a;
  }
}

extern "C" void kernel_launch(void* const* d_in, const int* in_sizes, int n_in,
                              void* d_out, int out_size, void* d_ws, size_t ws_size,
                              hipStream_t stream) {
  const float* pos = (const float*)d_in[0];
  const float* W1  = (const float*)d_in[1];
  const float* b1  = (const float*)d_in[2];
  const float* W2  = (const float*)d_in[3];
  const float* b2  = (const float*)d_in[4];
  const float* W3  = (const float*)d_in[5];
  // d_in[6] = b3: constant offset, no gradient -> unused.
  float* ws = (float*)d_ws;   // needs JSPLIT * 3072 floats = 96 KB

  dim3 grid(NPART / 16, JSPLIT);
  forces_kernel<<<grid, 256, 0, stream>>>(pos, W1, b1, W2, b2, W3, ws);
  reduce_kernel<<<(NPART * 3 + 255) / 256, 256, 0, stream>>>(ws, (float*)d_out);
}